// OriNet_23364622090937
// MI455X (gfx1250) — compile-verified
//
#include <hip/hip_runtime.h>
#include <hip/hip_bf16.h>

#define DD   512
#define DIM  1024
#define DS   16
#define DTRR 32
#define NLAY 8
#define KCV  4
#define BB   2
#define LLEN 1024
#define MROWS (BB*LLEN)       // 2048 token rows

typedef float v2f __attribute__((ext_vector_type(2)));
typedef float v8f __attribute__((ext_vector_type(8)));

__device__ __forceinline__ float siluf(float x)    { return x / (1.0f + __expf(-x)); }
__device__ __forceinline__ float softplusf(float x){ return (x > 20.0f) ? x : log1pf(__expf(x)); }

// ---------------- embedding gather: h[b,l,:] = embed[inputs[b,l],:] ----------------
__global__ __launch_bounds__(256) void embed_kernel(const int* __restrict__ idx,
                                                    const float* __restrict__ emb,
                                                    float* __restrict__ h) {
    int gid = blockIdx.x * blockDim.x + threadIdx.x;     // MROWS*DD
    int row = gid / DD, col = gid % DD;
    h[gid] = emb[(size_t)idx[row] * DD + col];
}

// ---------------- RMSNorm over rows of width 512 ----------------
__global__ __launch_bounds__(256) void rmsnorm_kernel(const float* __restrict__ in,
                                                      const float* __restrict__ w,
                                                      float* __restrict__ out) {
    __shared__ float red[256];
    __shared__ float s_scale;
    int row = blockIdx.x;
    int t   = threadIdx.x;
    const float* r = in + (size_t)row * DD;
    float a = r[t], b = r[t + 256];
    red[t] = a * a + b * b;
    __syncthreads();
    for (int s = 128; s > 0; s >>= 1) {
        if (t < s) red[t] += red[t + s];
        __syncthreads();
    }
    if (t == 0) s_scale = rsqrtf(red[0] * (1.0f / DD) + 1e-5f);
    __syncthreads();
    float scale = s_scale;
    out[(size_t)row * DD + t]       = a * scale * w[t];
    out[(size_t)row * DD + t + 256] = b * scale * w[t + 256];
}

// ---------------- fp32 WMMA GEMM: C[M,N] = A[M,K] @ W[N,K]^T ----------------
// One wave per 16x16 tile. mode: 0 = store, 1 = store softplus(acc + bias[n]), 2 = C += acc
__global__ __launch_bounds__(32) void gemm16_f32(const float* __restrict__ A, int lda,
                                                 const float* __restrict__ W, int ldw,
                                                 float* __restrict__ C, int ldc,
                                                 int K, const float* __restrict__ bias, int mode) {
    int lane = threadIdx.x;
    int half = lane >> 4;          // K-group select (lanes 16-31 hold K+2,K+3)
    int l16  = lane & 15;          // M (for A) / N (for B) within the tile
    int m0 = blockIdx.y << 4;
    int n0 = blockIdx.x << 4;

    const float* ap = A + (size_t)(m0 + l16) * lda + 2 * half;
    const float* wp = W + (size_t)(n0 + l16) * ldw + 2 * half;

    v8f acc = {0.f,0.f,0.f,0.f,0.f,0.f,0.f,0.f};
    for (int k = 0; k < K; k += 4) {
        v2f a = *(const v2f*)(ap + k);     // A[m][k+2h], A[m][k+2h+1]
        v2f b = *(const v2f*)(wp + k);     // B[k+2h][n] = W[n][k+2h]
        acc = __builtin_amdgcn_wmma_f32_16x16x4_f32(false, a, false, b,
                                                    (short)0, acc, false, false);
    }

    int col = n0 + l16;
    float bv = (mode == 1) ? bias[col] : 0.0f;
#pragma unroll
    for (int r = 0; r < 8; ++r) {
        int rowm = m0 + r + 8 * half;      // C/D layout: VGPR r -> M=r (+8 for hi lanes)
        float v  = acc[r];
        float* p = C + (size_t)rowm * ldc + col;
        if      (mode == 1) *p  = softplusf(v + bv);
        else if (mode == 2) *p += v;
        else                *p  = v;
    }
}

// ---------------- depthwise causal conv (both directions) + bias + SiLU ----------------
// xz rows are (2*DIM) wide; channel ch in [0,DIM). Backward dir = anti-causal window.
__global__ __launch_bounds__(256) void conv_silu_kernel(const float* __restrict__ xz,
                                                        const float* __restrict__ cw,
                                                        const float* __restrict__ cb,
                                                        float* __restrict__ xf,
                                                        float* __restrict__ xb) {
    int gid = blockIdx.x * blockDim.x + threadIdx.x;   // BB*LLEN*DIM
    int ch = gid % DIM;
    int t  = (gid / DIM) % LLEN;
    int b  = gid / (DIM * LLEN);
    float w0 = cw[ch*KCV+0], w1 = cw[ch*KCV+1], w2 = cw[ch*KCV+2], w3 = cw[ch*KCV+3];
    float bias = cb[ch];
    const size_t base = (size_t)b * LLEN;
    auto xval = [&](int tt) -> float {
        return (tt >= 0 && tt < LLEN) ? xz[(base + tt) * (2*DIM) + ch] : 0.0f;
    };
    float af = bias + w0*xval(t-3) + w1*xval(t-2) + w2*xval(t-1) + w3*xval(t);
    float ab = bias + w0*xval(t+3) + w1*xval(t+2) + w2*xval(t+1) + w3*xval(t);
    xf[gid] = siluf(af);
    xb[gid] = siluf(ab);
}

// ---------------- selective scan, both directions, one thread per (dir,b,channel) ----------------
__global__ __launch_bounds__(256) void scan_kernel(const float* __restrict__ xdbf,
                                                   const float* __restrict__ xdbb,
                                                   const float* __restrict__ dtf,
                                                   const float* __restrict__ dtb,
                                                   const float* __restrict__ xf,
                                                   const float* __restrict__ xb,
                                                   const float* __restrict__ A_log,
                                                   float* __restrict__ yf,
                                                   float* __restrict__ yb) {
    int tid = blockIdx.x * blockDim.x + threadIdx.x;   // 2*BB*DIM
    int dir = tid / (BB * DIM);
    int r   = tid % (BB * DIM);
    int b   = r / DIM, ch = r % DIM;

    const float* xdb = dir ? xdbb : xdbf;
    const float* dt  = dir ? dtb  : dtf;
    const float* x   = dir ? xb   : xf;
    float*       y   = dir ? yb   : yf;

    float a[DS];
#pragma unroll
    for (int s = 0; s < DS; ++s) a[s] = -__expf(A_log[(size_t)ch * DS + s]);
    float h[DS];
#pragma unroll
    for (int s = 0; s < DS; ++s) h[s] = 0.0f;

    for (int step = 0; step < LLEN; ++step) {
        int t   = dir ? (LLEN - 1 - step) : step;
        size_t row = (size_t)b * LLEN + t;
        float d  = dt[row * DIM + ch];
        float xv = x [row * DIM + ch];
        const float* Bp = xdb + row * (DTRR + 2*DS) + DTRR;   // B block
        const float* Cp = Bp + DS;                            // C block
        float dx = d * xv;
        float ys = 0.0f;
#pragma unroll
        for (int s = 0; s < DS; ++s) {
            float dA = __expf(d * a[s]);
            h[s] = dA * h[s] + dx * Bp[s];
            ys  += h[s] * Cp[s];
        }
        y[row * DIM + ch] = ys;
    }
}

// ---------------- gate: yf <- (yf + yb + (xf+xb)*Dp) * silu(z) ----------------
__global__ __launch_bounds__(256) void gate_kernel(const float* __restrict__ xz,
                                                   const float* __restrict__ xf,
                                                   const float* __restrict__ xb,
                                                   const float* __restrict__ yb,
                                                   const float* __restrict__ Dp,
                                                   float* __restrict__ yf) {
    int gid = blockIdx.x * blockDim.x + threadIdx.x;   // BB*LLEN*DIM
    int ch  = gid % DIM;
    int row = gid / DIM;
    float z = xz[(size_t)row * (2*DIM) + DIM + ch];
    float v = yf[gid] + yb[gid] + (xf[gid] + xb[gid]) * Dp[ch];
    yf[gid] = v * siluf(z);
}

// ---------------- head: logits[row,c] = sum_d hn[row,d]*hw[c,d], c in {0,1} ----------------
__global__ __launch_bounds__(256) void head_kernel(const float* __restrict__ hn,
                                                   const float* __restrict__ hw,
                                                   float* __restrict__ out) {
    int row = blockIdx.x * blockDim.x + threadIdx.x;   // MROWS
    const float* r = hn + (size_t)row * DD;
    float a0 = 0.f, a1 = 0.f;
    for (int d = 0; d < DD; ++d) {
        float v = r[d];
        a0 += v * hw[d];
        a1 += v * hw[DD + d];
    }
    out[row * 2 + 0] = a0;
    out[row * 2 + 1] = a1;
}

extern "C" void kernel_launch(void* const* d_in, const int* in_sizes, int n_in,
                              void* d_out, int out_size, void* d_ws, size_t ws_size,
                              hipStream_t stream) {
    const int*   inputs   = (const int*)  d_in[0];
    const float* embed    = (const float*)d_in[1];
    const float* Win      = (const float*)d_in[2];   // (NL, 2*DIM, DD)
    const float* conv_w   = (const float*)d_in[3];   // (NL, DIM, KC)
    const float* conv_b   = (const float*)d_in[4];   // (NL, DIM)
    const float* Wx       = (const float*)d_in[5];   // (NL, 64, DIM)
    const float* Wdt      = (const float*)d_in[6];   // (NL, DIM, 32)
    const float* bdt      = (const float*)d_in[7];   // (NL, DIM)
    const float* A_log    = (const float*)d_in[8];   // (NL, DIM, DS)
    const float* Dp       = (const float*)d_in[9];   // (NL, DIM)
    const float* Wout     = (const float*)d_in[10];  // (NL, DD, DIM)
    const float* norm_w   = (const float*)d_in[11];  // (NL, DD)
    const float* fnorm_w  = (const float*)d_in[12];  // (DD)
    const float* head_w   = (const float*)d_in[13];  // (2, DD)
    float* out = (float*)d_out;

    // ---- workspace carve (floats) ----
    float* ws = (float*)d_ws;
    size_t off = 0;
    float* h    = ws + off; off += (size_t)MROWS * DD;          // residual stream
    float* hn   = ws + off; off += (size_t)MROWS * DD;          // normed
    float* xz   = ws + off; off += (size_t)MROWS * 2 * DIM;     // Win output
    float* xf   = ws + off; off += (size_t)MROWS * DIM;         // conv+silu fwd
    float* xb   = ws + off; off += (size_t)MROWS * DIM;         // conv+silu bwd
    float* xdbf = ws + off; off += (size_t)MROWS * 64;          // [dtr|B|C] fwd
    float* xdbb = ws + off; off += (size_t)MROWS * 64;          // [dtr|B|C] bwd
    float* dtf  = ws + off; off += (size_t)MROWS * DIM;
    float* dtb  = ws + off; off += (size_t)MROWS * DIM;
    float* yf   = ws + off; off += (size_t)MROWS * DIM;         // reused as gated y
    float* yb   = ws + off; off += (size_t)MROWS * DIM;

    // ---- embedding ----
    embed_kernel<<<(MROWS * DD) / 256, 256, 0, stream>>>(inputs, embed, h);

    for (int l = 0; l < NLAY; ++l) {
        const float* Win_l  = Win    + (size_t)l * 2 * DIM * DD;
        const float* cw_l   = conv_w + (size_t)l * DIM * KCV;
        const float* cb_l   = conv_b + (size_t)l * DIM;
        const float* Wx_l   = Wx     + (size_t)l * 64 * DIM;
        const float* Wdt_l  = Wdt    + (size_t)l * DIM * DTRR;
        const float* bdt_l  = bdt    + (size_t)l * DIM;
        const float* Alog_l = A_log  + (size_t)l * DIM * DS;
        const float* Dp_l   = Dp     + (size_t)l * DIM;
        const float* Wout_l = Wout   + (size_t)l * DD * DIM;
        const float* nw_l   = norm_w + (size_t)l * DD;

        // hn = rmsnorm(h)
        rmsnorm_kernel<<<MROWS, 256, 0, stream>>>(h, nw_l, hn);

        // xz = hn @ Win^T   (2048 x 2048, K=512) — shared by both directions
        gemm16_f32<<<dim3(2*DIM/16, MROWS/16), 32, 0, stream>>>(
            hn, DD, Win_l, DD, xz, 2*DIM, DD, nullptr, 0);

        // depthwise conv + SiLU, both directions
        conv_silu_kernel<<<(MROWS * DIM) / 256, 256, 0, stream>>>(xz, cw_l, cb_l, xf, xb);

        // xdb = x @ Wx^T  (N=64, K=1024) per direction
        gemm16_f32<<<dim3(64/16, MROWS/16), 32, 0, stream>>>(xf, DIM, Wx_l, DIM, xdbf, 64, DIM, nullptr, 0);
        gemm16_f32<<<dim3(64/16, MROWS/16), 32, 0, stream>>>(xb, DIM, Wx_l, DIM, xdbb, 64, DIM, nullptr, 0);

        // dt = softplus(dtr @ Wdt^T + bdt)   (N=1024, K=32), dtr is cols 0..31 of xdb (lda=64)
        gemm16_f32<<<dim3(DIM/16, MROWS/16), 32, 0, stream>>>(xdbf, 64, Wdt_l, DTRR, dtf, DIM, DTRR, bdt_l, 1);
        gemm16_f32<<<dim3(DIM/16, MROWS/16), 32, 0, stream>>>(xdbb, 64, Wdt_l, DTRR, dtb, DIM, DTRR, bdt_l, 1);

        // selective scan, fwd + bwd in one launch (4096 channels)
        scan_kernel<<<(2 * BB * DIM) / 256, 256, 0, stream>>>(
            xdbf, xdbb, dtf, dtb, xf, xb, Alog_l, yf, yb);

        // gate: yf <- (yf + yb + (xf+xb)*Dp) * silu(z)
        gate_kernel<<<(MROWS * DIM) / 256, 256, 0, stream>>>(xz, xf, xb, yb, Dp_l, yf);

        // h += yf @ Wout^T   (N=512, K=1024), fused residual accumulate
        gemm16_f32<<<dim3(DD/16, MROWS/16), 32, 0, stream>>>(yf, DIM, Wout_l, DIM, h, DD, DIM, nullptr, 2);
    }

    // final norm + head
    rmsnorm_kernel<<<MROWS, 256, 0, stream>>>(h, fnorm_w, hn);
    head_kernel<<<MROWS / 256, 256, 0, stream>>>(hn, head_w, out);
}